// ExpertWeightsFp8E4M3FN_84713934946388
// MI455X (gfx1250) — compile-verified
//
#include <hip/hip_runtime.h>
#include <hip/hip_bf16.h>
#include <stdint.h>

typedef uint32_t u32;
typedef u32   v4u  __attribute__((ext_vector_type(4)));
typedef float v2f  __attribute__((ext_vector_type(2)));

// ---------------------------------------------------------------------------
// E=64, O=4096, I=1024. out[e,o,i] = bf16(fp8_e4m3fn(w)) * bf16(scale[e,o]).
// 16 fp8 elements per thread: one b128 NT load -> two b128 NT stores.
// I/16 = 64 chunks per (e,o) row -> row = chunk_id >> 6 (wave-uniform scale).
// ---------------------------------------------------------------------------

#if __has_builtin(__builtin_amdgcn_cvt_pk_f32_fp8)
#define USE_HW_FP8_CVT 1
#else
#define USE_HW_FP8_CVT 0
#endif

// Packed bf16 multiply: D.lo = a.lo*b.lo, D.hi = a.hi*b.hi (RNE) — VOP3P op 42.
__device__ __forceinline__ u32 pk_mul_bf16(u32 a, u32 b) {
    u32 d;
    asm("v_pk_mul_bf16 %0, %1, %2" : "=v"(d) : "v"(a), "v"(b));
    return d;
}

// Exact truncation-pack of two fp8-exact f32 values into bf16x2.
// (fp8 e4m3fn -> bf16 is lossless, so plain truncation is bit-exact.)
// Backend folds this OR-of-byte-extracts into v_perm_b32 itself.
__device__ __forceinline__ u32 pack_bf16x2_trunc(float lo, float hi) {
    u32 a = __builtin_bit_cast(u32, lo);
    u32 b = __builtin_bit_cast(u32, hi);
    return (a >> 16) | (b & 0xFFFF0000u);
}

#if !USE_HW_FP8_CVT
// Fallback: exact integer/float decode of one fp8 e4m3fn byte -> f32.
__device__ __forceinline__ float fp8_to_f32_sw(u32 b) {
    u32 mag = b & 0x7Fu;                       // exp[6:3], mant[2:0]
    // Place exp/mant into f32 fields; value = bits_as_f32 * 2^120 (denorm-safe).
    float f = __builtin_bit_cast(float, mag << 20) * 0x1p+120f;
    u32 fb  = __builtin_bit_cast(u32, f) | ((b & 0x80u) << 24);
    return __builtin_bit_cast(float, fb);
}
#endif

struct BfPair { u32 d01, d23; };

// Convert 4 packed fp8 bytes (one dword) into two bf16x2 dwords, scaled.
__device__ __forceinline__ BfPair cvt4_scaled(u32 w, u32 s2) {
#if USE_HW_FP8_CVT
    v2f f01 = __builtin_amdgcn_cvt_pk_f32_fp8((int)w, false);  // bytes 0,1
    v2f f23 = __builtin_amdgcn_cvt_pk_f32_fp8((int)w, true);   // bytes 2,3
    u32 b01 = pack_bf16x2_trunc(f01.x, f01.y);
    u32 b23 = pack_bf16x2_trunc(f23.x, f23.y);
#else
    u32 b01 = pack_bf16x2_trunc(fp8_to_f32_sw(w & 0xFF),
                                fp8_to_f32_sw((w >> 8) & 0xFF));
    u32 b23 = pack_bf16x2_trunc(fp8_to_f32_sw((w >> 16) & 0xFF),
                                fp8_to_f32_sw((w >> 24) & 0xFF));
#endif
    BfPair r;
    r.d01 = pk_mul_bf16(b01, s2);
    r.d23 = pk_mul_bf16(b23, s2);
    return r;
}

__global__ void __launch_bounds__(256)
expert_dequant_fp8_bf16(const v4u* __restrict__ w16,     // 16 fp8 per v4u
                        const float* __restrict__ scale, // (E*O) f32
                        v4u* __restrict__ out,           // 8 bf16 per v4u
                        u32 n_chunks)                    // total 16-elem chunks
{
    const u32 tid = blockIdx.x * 256u + threadIdx.x;
    if (tid >= n_chunks) return;

    const u32 row = tid >> 6;                  // 64 chunks per 1024-elem row
    // scale -> bf16 (RNE, matches scale.astype(bf16)), broadcast to both halves
    u32 us = __builtin_bit_cast(u32, scale[row]);
    us += 0x7FFFu + ((us >> 16) & 1u);
    const u32 s2 = (us >> 16) | (us & 0xFFFF0000u);

    // Streaming data: non-temporal so 768 MB doesn't thrash the 192 MB L2.
    const v4u wv = __builtin_nontemporal_load(w16 + tid);

    BfPair p0 = cvt4_scaled(wv.x, s2);
    BfPair p1 = cvt4_scaled(wv.y, s2);
    BfPair p2 = cvt4_scaled(wv.z, s2);
    BfPair p3 = cvt4_scaled(wv.w, s2);

    v4u o0, o1;
    o0.x = p0.d01; o0.y = p0.d23; o0.z = p1.d01; o0.w = p1.d23;
    o1.x = p2.d01; o1.y = p2.d23; o1.z = p3.d01; o1.w = p3.d23;

    __builtin_nontemporal_store(o0, out + 2u * tid);
    __builtin_nontemporal_store(o1, out + 2u * tid + 1u);
}

extern "C" void kernel_launch(void* const* d_in, const int* in_sizes, int n_in,
                              void* d_out, int out_size, void* d_ws, size_t ws_size,
                              hipStream_t stream) {
    const v4u*   w     = (const v4u*)d_in[0];    // uint8 fp8 bit patterns
    const float* scale = (const float*)d_in[1];  // f32 (E*O)
    v4u*         out   = (v4u*)d_out;            // bf16 output, viewed 8-at-a-time

    const u32 n_chunks = (u32)(in_sizes[0] / 16); // 16 fp8 elems per thread
    const u32 blocks   = (n_chunks + 255u) / 256u;

    expert_dequant_fp8_bf16<<<dim3(blocks), dim3(256), 0, stream>>>(w, scale, out, n_chunks);
}